// PointPillarScatter_11974368821926
// MI455X (gfx1250) — compile-verified
//
#include <hip/hip_runtime.h>

// PointPillarScatter for MI455X (gfx1250).
// Output layout: [B, C=64, NY=496, NX=432] float32.
// Kernel 1: NT-streamed zero-fill of the canvas (b128 stores).
// Kernel 2: async-stage pillar feature rows to LDS, then scatter.

#define PPS_NX 432
#define PPS_NY 496
#define PPS_CH 64
#define PPS_PPB 4   // pillars per block (256 threads = 4 pillars x 64 channels)

typedef float v4f __attribute__((ext_vector_type(4)));
typedef int   v4i __attribute__((vector_size(16)));
typedef __attribute__((address_space(1))) v4i* v4i_glob;
typedef __attribute__((address_space(3))) v4i* v4i_lds;

#if __has_builtin(__builtin_amdgcn_global_load_async_to_lds_b128) && \
    __has_builtin(__builtin_amdgcn_s_wait_asynccnt)
#define PPS_USE_ASYNC_LDS 1
#else
#define PPS_USE_ASYNC_LDS 0
#endif

// ---------------------------------------------------------------------------
// Kernel 1: zero the canvas with non-temporal 128-bit streaming stores.
// ---------------------------------------------------------------------------
__global__ void pps_zero_canvas(float* __restrict__ out, long long n4, long long ntot) {
    const long long stride = (long long)gridDim.x * blockDim.x;
    long long i = (long long)blockIdx.x * blockDim.x + threadIdx.x;
    v4f z = {0.0f, 0.0f, 0.0f, 0.0f};
    v4f* __restrict__ out4 = (v4f*)out;
    for (; i < n4; i += stride) {
        __builtin_nontemporal_store(z, out4 + i);   // global_store_b128 th:NT
    }
    // Scalar tail (out_size not divisible by 4 — not hit for this shape, but safe).
    if (blockIdx.x == 0 && threadIdx.x == 0) {
        for (long long r = n4 * 4; r < ntot; ++r) out[r] = 0.0f;
    }
}

// ---------------------------------------------------------------------------
// Kernel 2: scatter pillar feature rows into the canvas.
//   - 256 threads/block = PPS_PPB pillars x 64 channels
//   - coalesced 1KB feature read per block, staged through LDS via the
//     gfx1250 async copy engine when available (ASYNCcnt-tracked)
//   - one scattered 4B store per thread (irreducible, channel-major output)
// ---------------------------------------------------------------------------
__global__ void pps_scatter(const float* __restrict__ feat,
                            const int*  __restrict__ coords,
                            float*      __restrict__ out,
                            int P) {
    __shared__ float tile[PPS_PPB * PPS_CH];

    const int t     = threadIdx.x;              // 0..255
    const int pBase = blockIdx.x * PPS_PPB;
    const long long totalElems = (long long)P * PPS_CH;

#if PPS_USE_ASYNC_LDS
    // Threads 0..63 (waves 0,1) each issue one 16B async global->LDS copy.
    if (t < (PPS_PPB * PPS_CH) / 4) {
        const long long e = (long long)pBase * PPS_CH + (long long)t * 4;
        if (e + 3 < totalElems) {
            __builtin_amdgcn_global_load_async_to_lds_b128(
                (v4i_glob)(feat + e),
                (v4i_lds)(&tile[t * 4]),
                /*offset=*/0, /*cpol=*/0);
        }
    }
    // Each issuing wave drains its own ASYNCcnt before signaling the barrier,
    // so after the barrier the LDS tile is visible to all 8 waves.
    __builtin_amdgcn_s_wait_asynccnt(0);
    __syncthreads();
#else
    {
        const long long e = (long long)pBase * PPS_CH + t;
        tile[t] = (e < totalElems) ? feat[e] : 0.0f;
    }
    __syncthreads();
#endif

    const int q = t >> 6;          // pillar within block
    const int c = t & (PPS_CH - 1);
    const int p = pBase + q;
    if (p < P) {
        const int b = coords[4 * p + 0];
        const int y = coords[4 * p + 2];
        const int x = coords[4 * p + 3];
        const long long idx =
            ((((long long)b * PPS_CH + c) * PPS_NY) + y) * PPS_NX + x;
        out[idx] = tile[t];
    }
}

// ---------------------------------------------------------------------------
// Launch
// ---------------------------------------------------------------------------
extern "C" void kernel_launch(void* const* d_in, const int* in_sizes, int n_in,
                              void* d_out, int out_size, void* d_ws, size_t ws_size,
                              hipStream_t stream) {
    const float* feat   = (const float*)d_in[0];  // [P, 64] f32
    const int*   coords = (const int*)d_in[1];    // [P, 4]  (b, z, y, x)
    float*       out    = (float*)d_out;          // [B, 64, 496, 432] f32

    const int P = in_sizes[0] / PPS_CH;

    // 1) zero-fill canvas
    const long long ntot = (long long)out_size;
    const long long n4   = ntot / 4;
    long long zb = (n4 + 255) / 256;
    if (zb > 131072) zb = 131072;               // grid-stride covers the rest
    if (zb < 1)      zb = 1;
    pps_zero_canvas<<<(unsigned)zb, 256, 0, stream>>>(out, n4, ntot);

    // 2) scatter pillars (same stream -> ordered after zero-fill)
    const int sblocks = (P + PPS_PPB - 1) / PPS_PPB;
    if (sblocks > 0) {
        pps_scatter<<<sblocks, 256, 0, stream>>>(feat, coords, out, P);
    }
}